// SemanticModule_38603166057109
// MI455X (gfx1250) — compile-verified
//
#include <hip/hip_runtime.h>

typedef float v2f __attribute__((ext_vector_type(2)));
typedef float v8f __attribute__((ext_vector_type(8)));

// ---------------------------------------------------------------------------
// [100000 x 32] @ [32 x 32] + bias  -> Y, via V_WMMA_F32_16X16X4_F32.
// One wave computes one 16x16 tile of Y; K=32 -> 8 accumulating WMMAs.
// ---------------------------------------------------------------------------
__global__ __launch_bounds__(256) void gemm32_wmma_kernel(
    const float* __restrict__ H, const float* __restrict__ W,
    const float* __restrict__ bias, float* __restrict__ Y, int nrows) {
  __shared__ float sW[32 * 32];
  __shared__ float sB[32];
  for (int t = threadIdx.x; t < 1024; t += 256) sW[t] = W[t];
  if (threadIdx.x < 32) sB[threadIdx.x] = bias[threadIdx.x];
  __syncthreads();

  const int wave = threadIdx.x >> 5;
  const int lane = threadIdx.x & 31;
  const int tile = blockIdx.x * 8 + wave;
  const int mt = tile >> 1;       // 16-row tile index
  const int nt = tile & 1;        // 16-col tile index (32 cols -> 2 tiles)
  if (mt * 16 >= nrows) return;   // uniform per wave; EXEC stays all-ones

  const int m15   = lane & 15;
  const int khalf = (lane >> 4) << 1;           // lanes 16-31 carry K+2/K+3
  const int row   = mt * 16 + m15;              // A: M = lane (both halves)
  const int col   = nt * 16 + m15;              // B/D: N = lane & 15

  const float* hrow = H + (size_t)row * 32 + khalf;

  v8f acc = {};
#pragma unroll
  for (int k0 = 0; k0 < 32; k0 += 4) {
    v2f a;
    a.x = hrow[k0];                              // A[M=row, K=k0+khalf]
    a.y = hrow[k0 + 1];                          // A[M=row, K=k0+khalf+1]
    v2f b;
    b.x = sW[(k0 + khalf) * 32 + col];           // B[K, N=col]
    b.y = sW[(k0 + khalf + 1) * 32 + col];
    acc = __builtin_amdgcn_wmma_f32_16x16x4_f32(
        /*neg_a=*/false, a, /*neg_b=*/false, b,
        /*c_mod=*/(short)0, acc, /*reuse_a=*/false, /*reuse_b=*/false);
  }

  // D layout: VGPR j -> M = j (lanes 0-15) or j+8 (lanes 16-31), N = lane&15
  const int rbase = mt * 16 + ((lane >> 4) << 3);
  const float bc = sB[col];
  float* out = Y + (size_t)rbase * 32 + col;
#pragma unroll
  for (int j = 0; j < 8; ++j) out[(size_t)j * 32] = acc[j] + bc;
}

// ---------------------------------------------------------------------------
// Edge scatter: out[dst] += Y[src] * (inv ? inv[dst] : 1). 8 threads/edge,
// float4 gather per thread, 4 f32 atomics (L2-resident destination).
// ---------------------------------------------------------------------------
__global__ void scatter_kernel(const float* __restrict__ Y,
                               const int* __restrict__ src,
                               const int* __restrict__ dst,
                               const float* __restrict__ inv,
                               float* __restrict__ out, int ne) {
  long long t = (long long)blockIdx.x * blockDim.x + threadIdx.x;
  int e = (int)(t >> 3);
  if (e >= ne) return;
  int c = (int)(t & 7) << 2;
  int s = src[e], d = dst[e];
  float scale = inv ? inv[d] : 1.0f;
  const float4 v = *(const float4*)(Y + (size_t)s * 32 + c);
  float* o = out + (size_t)d * 32 + c;
  atomicAdd(o + 0, v.x * scale);
  atomicAdd(o + 1, v.y * scale);
  atomicAdd(o + 2, v.z * scale);
  atomicAdd(o + 3, v.w * scale);
}

// ---------------------------------------------------------------------------
// Head (D_IN = 6): self term summed over 3 relations, and per-relation msg.
// ---------------------------------------------------------------------------
__global__ void head_self_kernel(const float* __restrict__ x,
                                 const float* __restrict__ Ws,
                                 const float* __restrict__ bs,
                                 float* __restrict__ h, int n) {
  int t = blockIdx.x * blockDim.x + threadIdx.x;
  if (t >= n * 32) return;
  int i = t >> 5, c = t & 31;
  float acc = bs[c] + bs[32 + c] + bs[64 + c];
#pragma unroll
  for (int j = 0; j < 6; ++j) {
    float w = Ws[j * 32 + c] + Ws[192 + j * 32 + c] + Ws[384 + j * 32 + c];
    acc += x[i * 6 + j] * w;
  }
  h[t] = acc;
}

__global__ void head_msg_kernel(const float* __restrict__ x,
                                const float* __restrict__ Wm,
                                const float* __restrict__ bm,
                                float* __restrict__ Y, int n) {
  int t = blockIdx.x * blockDim.x + threadIdx.x;
  if (t >= n * 32) return;
  int i = t >> 5, c = t & 31;
  float acc = bm[c];
#pragma unroll
  for (int j = 0; j < 6; ++j) acc += x[i * 6 + j] * Wm[j * 32 + c];
  Y[t] = acc;
}

// ---------------------------------------------------------------------------
// Degree counts / elementwise helpers.
// ---------------------------------------------------------------------------
__global__ void zero_kernel(float* __restrict__ p, int n) {
  int t = blockIdx.x * blockDim.x + threadIdx.x;
  if (t < n) p[t] = 0.0f;
}
__global__ void count_kernel(const int* __restrict__ dst, float* __restrict__ cnt, int ne) {
  int t = blockIdx.x * blockDim.x + threadIdx.x;
  if (t < ne) atomicAdd(cnt + dst[t], 1.0f);
}
__global__ void inv_kernel(float* __restrict__ p, int n) {
  int t = blockIdx.x * blockDim.x + threadIdx.x;
  if (t < n) p[t] = 1.0f / fmaxf(p[t], 1.0f);
}
__global__ void scale4_kernel(const float* __restrict__ in, float* __restrict__ out, int n) {
  int t = blockIdx.x * blockDim.x + threadIdx.x;
  if (t < n) out[t] = 4.0f * in[t];   // layers: 3x identity self + residual
}
__global__ void relu_kernel(const float* __restrict__ in, float* __restrict__ out, int n) {
  int t = blockIdx.x * blockDim.x + threadIdx.x;
  if (t < n) out[t] = fmaxf(in[t], 0.0f);
}

static inline unsigned gridFor(long long threads, int block) {
  return (unsigned)((threads + block - 1) / block);
}

extern "C" void kernel_launch(void* const* d_in, const int* in_sizes, int n_in,
                              void* d_out, int out_size, void* d_ws, size_t ws_size,
                              hipStream_t stream) {
  const float* x         = (const float*)d_in[0];
  const int*   ei0       = (const int*)d_in[1];   // intersects      (mean)
  const int*   ei1       = (const int*)d_in[2];   // temp_previous   (add)
  const int*   ei2       = (const int*)d_in[3];   // represented_by  (mean)
  const float* Wm_head   = (const float*)d_in[4]; // [3][6][32]
  const float* bm_head   = (const float*)d_in[5]; // [3][32]
  const float* Ws_head   = (const float*)d_in[6]; // [3][6][32]
  const float* bs_head   = (const float*)d_in[7]; // [3][32]
  const float* Wm_layers = (const float*)d_in[8]; // [4][3][32][32]
  const float* bm_layers = (const float*)d_in[9]; // [4][3][32]

  const int n  = in_sizes[0] / 6;
  const int e0 = in_sizes[1] / 2;
  const int e1 = in_sizes[2] / 2;
  const int e2 = in_sizes[3] / 2;

  float* ws   = (float*)d_ws;
  float* h0   = ws;                         // [n*32]
  float* h1   = ws + (size_t)n * 32;        // [n*32]
  float* Y    = ws + (size_t)n * 64;        // [n*32] per-relation messages
  float* inv0 = ws + (size_t)n * 96;        // [n] 1/deg for relation 0
  float* inv2 = inv0 + n;                   // [n] 1/deg for relation 2

  const int*   srcs[3] = {ei0, ei1, ei2};
  const int*   dsts[3] = {ei0 + e0, ei1 + e1, ei2 + e2};
  const int    nes[3]  = {e0, e1, e2};
  const float* invs[3] = {inv0, nullptr, inv2};   // ("mean","add","mean")

  const int B = 256;
  const int nc = n * 32;

  // --- degree -> inverse-degree (relations 0 and 2) ------------------------
  zero_kernel<<<gridFor(2 * n, B), B, 0, stream>>>(inv0, 2 * n);
  count_kernel<<<gridFor(e0, B), B, 0, stream>>>(dsts[0], inv0, e0);
  count_kernel<<<gridFor(e2, B), B, 0, stream>>>(dsts[2], inv2, e2);
  inv_kernel<<<gridFor(2 * n, B), B, 0, stream>>>(inv0, 2 * n);

  // --- head conv: h0 = sum_r [ agg_r(x@Wm_r + bm_r) + x@Ws_r + bs_r ] ------
  head_self_kernel<<<gridFor(nc, B), B, 0, stream>>>(x, Ws_head, bs_head, h0, n);
  for (int r = 0; r < 3; ++r) {
    head_msg_kernel<<<gridFor(nc, B), B, 0, stream>>>(
        x, Wm_head + r * 192, bm_head + r * 32, Y, n);
    scatter_kernel<<<gridFor((long long)nes[r] * 8, B), B, 0, stream>>>(
        Y, srcs[r], dsts[r], invs[r], h0, nes[r]);
  }

  // --- 4 residual layers: h_new = sum_r agg_r(h@Wm_lr + bm_lr) + 4h --------
  const unsigned gemmBlocks = gridFor((long long)((n + 15) / 16) * 2, 8);
  float* hin = h0;
  float* hout = h1;
  for (int l = 0; l < 4; ++l) {
    scale4_kernel<<<gridFor(nc, B), B, 0, stream>>>(hin, hout, nc);
    for (int r = 0; r < 3; ++r) {
      gemm32_wmma_kernel<<<gemmBlocks, B, 0, stream>>>(
          hin, Wm_layers + (size_t)(l * 3 + r) * 1024,
          bm_layers + (size_t)(l * 3 + r) * 32, Y, n);
      scatter_kernel<<<gridFor((long long)nes[r] * 8, B), B, 0, stream>>>(
          Y, srcs[r], dsts[r], invs[r], hout, nes[r]);
    }
    float* tmp = hin; hin = hout; hout = tmp;
  }

  // --- final ReLU into d_out ----------------------------------------------
  relu_kernel<<<gridFor(nc, B), B, 0, stream>>>(hin, (float*)d_out, nc);
}